// HANLayer_36670430773760
// MI455X (gfx1250) — compile-verified
//
#include <hip/hip_runtime.h>
#include <hip/hip_bf16.h>

#define N_NODES 50000
#define W_IN    256
#define W_OUT   256
#define HEADS   8
#define D_HEAD  32
#define NUM_EDGE 3
#define N_EDGES 800000
#define NEG_SLOPE 0.2f

typedef __attribute__((ext_vector_type(2))) float v2f;
typedef __attribute__((ext_vector_type(8))) float v8f;

// ---------- float atomic helpers ----------
__device__ __forceinline__ float atomAddF(float* p, float v) {
    return __hip_atomic_fetch_add(p, v, __ATOMIC_RELAXED, __HIP_MEMORY_SCOPE_AGENT);
}
// works for mixed-sign floats when slot is initialized to -inf
__device__ __forceinline__ void atomMaxF(float* p, float v) {
    if (v >= 0.0f) atomicMax((int*)p, __float_as_int(v));
    else           atomicMin((unsigned int*)p, __float_as_uint(v));
}

// ---------- 1) h_t = x @ W_t via V_WMMA_F32_16X16X4_F32 ----------
// grid (3125, 1, 3); block 128 (4 waves). Wave w computes the 16x64 tile at
// rows bx*16, cols w*64: one shared A fragment feeds 4 independent
// accumulators per k-step (4 WMMA dep-chains, A load amortized 4x).
__global__ void gemm_wmma(const float* __restrict__ x,
                          const float* __restrict__ Ws,
                          float* __restrict__ h) {
    const int t    = blockIdx.z;
    const int wave = threadIdx.x >> 5;
    const int lane = threadIdx.x & 31;

    const float* W = Ws + (size_t)t * W_IN * W_OUT;
    float*       H = h  + (size_t)t * N_NODES * W_OUT;

    const int row0 = blockIdx.x * 16;
    const int col0 = wave * 64;

    const int mrow = lane & 15;         // M index for A-frag / N index for B-frag
    const int kk   = (lane >> 4) * 2;   // lanes 16-31 carry K=2,3 of each step

    const float* arow = x + (size_t)(row0 + mrow) * W_IN;

    v8f acc0 = {}, acc1 = {}, acc2 = {}, acc3 = {};
    #pragma unroll 2
    for (int k = 0; k < W_IN; k += 4) {
        v2f a;
        a.x = arow[k + kk];
        a.y = arow[k + kk + 1];
        const float* wp = W + (size_t)(k + kk) * W_OUT + col0 + mrow;
        v2f b0, b1, b2, b3;
        b0.x = wp[0];           b0.y = wp[W_OUT];
        b1.x = wp[16];          b1.y = wp[W_OUT + 16];
        b2.x = wp[32];          b2.y = wp[W_OUT + 32];
        b3.x = wp[48];          b3.y = wp[W_OUT + 48];
        acc0 = __builtin_amdgcn_wmma_f32_16x16x4_f32(false, a, false, b0,
                                                     (short)0, acc0, false, false);
        acc1 = __builtin_amdgcn_wmma_f32_16x16x4_f32(false, a, false, b1,
                                                     (short)0, acc1, false, false);
        acc2 = __builtin_amdgcn_wmma_f32_16x16x4_f32(false, a, false, b2,
                                                     (short)0, acc2, false, false);
        acc3 = __builtin_amdgcn_wmma_f32_16x16x4_f32(false, a, false, b3,
                                                     (short)0, acc3, false, false);
    }

    // D layout: VGPR r -> row (lane<16 ? r : 8+r), col = lane&15
    const int crow = row0 + (lane >> 4) * 8;
    float* Hp = H + (size_t)crow * W_OUT + col0 + (lane & 15);
    #pragma unroll
    for (int r = 0; r < 8; ++r) {
        Hp[(size_t)r * W_OUT +  0] = acc0[r];
        Hp[(size_t)r * W_OUT + 16] = acc1[r];
        Hp[(size_t)r * W_OUT + 32] = acc2[r];
        Hp[(size_t)r * W_OUT + 48] = acc3[r];
    }
}

// ---------- 2) el/er per (node, head); init m=-inf, s=0, out=bias ----------
// grid (N_NODES, 3); block 256; wave w handles head w (wave32 shuffle reduce).
__global__ void scores_init(const float* __restrict__ h,
                            const float* __restrict__ a_l,
                            const float* __restrict__ a_r,
                            const float* __restrict__ bias,
                            float* __restrict__ el, float* __restrict__ er,
                            float* __restrict__ mb, float* __restrict__ sb,
                            float* __restrict__ oacc) {
    const int n = blockIdx.x;
    const int t = blockIdx.y;
    const int c = threadIdx.x;           // head*32 + d
    const int head = c >> 5;
    const int d    = c & 31;

    const size_t nb = ((size_t)t * N_NODES + n) * W_OUT;
    const float hv = h[nb + c];
    float pl = hv * a_l[t * W_OUT + c];
    float pr = hv * a_r[t * W_OUT + c];
    #pragma unroll
    for (int off = 16; off > 0; off >>= 1) {
        pl += __shfl_xor(pl, off, 32);
        pr += __shfl_xor(pr, off, 32);
    }
    const size_t sbase = ((size_t)t * N_NODES + n) * HEADS;
    if (d == 0) {
        el[sbase + head] = pl;
        er[sbase + head] = pr;
        mb[sbase + head] = -INFINITY;
        sb[sbase + head] = 0.0f;
    }
    oacc[nb + c] = bias[t * W_OUT + c];
}

// ---------- 3) segment max over incoming edges ----------
// grid (25000, 1, 3); block 256; thread per (edge, head)
__global__ void edge_max(const long long* __restrict__ adj,
                         const float* __restrict__ el,
                         const float* __restrict__ er,
                         float* __restrict__ mb) {
    const int t = blockIdx.z;
    const long long idx = (long long)blockIdx.x * blockDim.x + threadIdx.x;
    if (idx >= (long long)N_EDGES * HEADS) return;
    const int edge = (int)(idx >> 3);
    const int head = (int)(idx & 7);
    const long long* A = adj + (size_t)t * 2 * N_EDGES;
    const int src = (int)A[edge];
    const int dst = (int)A[N_EDGES + edge];
    float e = el[((size_t)t * N_NODES + src) * HEADS + head]
            + er[((size_t)t * N_NODES + dst) * HEADS + head];
    e = (e >= 0.0f) ? e : NEG_SLOPE * e;
    atomMaxF(&mb[((size_t)t * N_NODES + dst) * HEADS + head], e);
}

// ---------- 4) segment sum of exp(e - m[dst]) ----------
__global__ void edge_sum(const long long* __restrict__ adj,
                         const float* __restrict__ el,
                         const float* __restrict__ er,
                         const float* __restrict__ mb,
                         float* __restrict__ sb) {
    const int t = blockIdx.z;
    const long long idx = (long long)blockIdx.x * blockDim.x + threadIdx.x;
    if (idx >= (long long)N_EDGES * HEADS) return;
    const int edge = (int)(idx >> 3);
    const int head = (int)(idx & 7);
    const long long* A = adj + (size_t)t * 2 * N_EDGES;
    const int src = (int)A[edge];
    const int dst = (int)A[N_EDGES + edge];
    float e = el[((size_t)t * N_NODES + src) * HEADS + head]
            + er[((size_t)t * N_NODES + dst) * HEADS + head];
    e = (e >= 0.0f) ? e : NEG_SLOPE * e;
    const size_t db = ((size_t)t * N_NODES + dst) * HEADS + head;
    atomAddF(&sb[db], __expf(e - mb[db]));
}

// ---------- 5) out[dst] += alpha * h[src] ----------
// grid (N_EDGES, 1, 3); block 256 (one edge per block, thread per channel)
__global__ void edge_scatter(const long long* __restrict__ adj,
                             const float* __restrict__ h,
                             const float* __restrict__ el,
                             const float* __restrict__ er,
                             const float* __restrict__ mb,
                             const float* __restrict__ sb,
                             float* __restrict__ oacc) {
    const int t = blockIdx.z;
    const int edge = blockIdx.x;
    const int c = threadIdx.x;
    const int head = c >> 5;
    const long long* A = adj + (size_t)t * 2 * N_EDGES;
    const int src = (int)A[edge];
    const int dst = (int)A[N_EDGES + edge];
    const size_t db = ((size_t)t * N_NODES + dst) * HEADS + head;
    float e = el[((size_t)t * N_NODES + src) * HEADS + head] + er[db];
    e = (e >= 0.0f) ? e : NEG_SLOPE * e;
    e = __expf(e - mb[db]);
    const float alpha = e / (sb[db] + 1e-9f);
    const float hv = h[((size_t)t * N_NODES + src) * W_OUT + c];
    atomAddF(&oacc[((size_t)t * N_NODES + dst) * W_OUT + c], alpha * hv);
}

// ---------- 6) semantic attention combine (no softmax, faithful) ----------
// grid (N_NODES); block 256
__global__ void semantic_att(const float* __restrict__ oacc,
                             const float* __restrict__ att_w,
                             const float* __restrict__ att_b,
                             float* __restrict__ out) {
    __shared__ float red[3 * 8];
    const int n = blockIdx.x;
    const int c = threadIdx.x;
    const int wave = c >> 5;
    const int lane = c & 31;

    const float v0 = oacc[((size_t)0 * N_NODES + n) * W_OUT + c];
    const float v1 = oacc[((size_t)1 * N_NODES + n) * W_OUT + c];
    const float v2 = oacc[((size_t)2 * N_NODES + n) * W_OUT + c];
    const float aw = att_w[c];
    float p0 = v0 * aw, p1 = v1 * aw, p2 = v2 * aw;
    #pragma unroll
    for (int off = 16; off > 0; off >>= 1) {
        p0 += __shfl_xor(p0, off, 32);
        p1 += __shfl_xor(p1, off, 32);
        p2 += __shfl_xor(p2, off, 32);
    }
    if (lane == 0) { red[wave] = p0; red[8 + wave] = p1; red[16 + wave] = p2; }
    __syncthreads();
    float a0 = att_b[0], a1 = att_b[0], a2 = att_b[0];
    #pragma unroll
    for (int i = 0; i < 8; ++i) { a0 += red[i]; a1 += red[8 + i]; a2 += red[16 + i]; }
    out[(size_t)n * W_OUT + c] = a0 * v0 + a1 * v1 + a2 * v2;
}

extern "C" void kernel_launch(void* const* d_in, const int* in_sizes, int n_in,
                              void* d_out, int out_size, void* d_ws, size_t ws_size,
                              hipStream_t stream) {
    const float*     x     = (const float*)d_in[0];
    const long long* adj   = (const long long*)d_in[1];   // int64 per reference
    const float*     Ws    = (const float*)d_in[2];
    const float*     a_l   = (const float*)d_in[3];
    const float*     a_r   = (const float*)d_in[4];
    const float*     bias  = (const float*)d_in[5];
    const float*     att_w = (const float*)d_in[6];
    const float*     att_b = (const float*)d_in[7];
    float*           out   = (float*)d_out;

    // workspace layout (floats)
    float* ws   = (float*)d_ws;
    const size_t HN = (size_t)NUM_EDGE * N_NODES;           // 150000
    float* h    = ws;                                        // HN*256
    float* oacc = h    + HN * W_OUT;                         // HN*256
    float* el   = oacc + HN * W_OUT;                         // HN*8
    float* er   = el   + HN * HEADS;
    float* mb   = er   + HN * HEADS;
    float* sb   = mb   + HN * HEADS;

    // 1) WMMA GEMMs: h_t = x @ W_t  (wave computes 16x64, 4 acc chains)
    gemm_wmma<<<dim3(N_NODES / 16, 1, NUM_EDGE), 128, 0, stream>>>(x, Ws, h);

    // 2) per-node scores + init accumulators
    scores_init<<<dim3(N_NODES, NUM_EDGE), 256, 0, stream>>>(
        h, a_l, a_r, bias, el, er, mb, sb, oacc);

    // 3-4) segment softmax statistics over edges
    const int eh_blocks = (N_EDGES * HEADS + 255) / 256;    // 25000
    edge_max<<<dim3(eh_blocks, 1, NUM_EDGE), 256, 0, stream>>>(adj, el, er, mb);
    edge_sum<<<dim3(eh_blocks, 1, NUM_EDGE), 256, 0, stream>>>(adj, el, er, mb, sb);

    // 5) weighted scatter-aggregate
    edge_scatter<<<dim3(N_EDGES, 1, NUM_EDGE), 256, 0, stream>>>(
        adj, h, el, er, mb, sb, oacc);

    // 6) semantic attention combine
    semantic_att<<<dim3(N_NODES), 256, 0, stream>>>(oacc, att_w, att_b, out);
}